// CTCLayer_58634893525673
// MI455X (gfx1250) — compile-verified
//
#include <hip/hip_runtime.h>
#include <stdint.h>

static constexpr int B_  = 64;
static constexpr int T_  = 2048;
static constexpr int C_  = 256;
static constexpr int L_  = 256;
static constexpr int S_  = 513;      // 2*L+1
static constexpr int BLANK_ = 255;
static constexpr int DEPTH  = 8;     // prefetch distance (timesteps in flight)
static constexpr int RING   = DEPTH + 1; // 9 slots: refill target is last step's slot
#define NEGV (-1e30f)

// ---- fast hw transcendentals (v_log_f32 / v_exp_f32 are base-2) ----
__device__ __forceinline__ float fast_ln(float x) {
  return __builtin_amdgcn_logf(x) * 0.6931471805599453f;
}
__device__ __forceinline__ float fast_exp(float x) {
  return __builtin_amdgcn_exp2f(x * 1.4426950408889634f);
}

// ---- async global -> LDS copy, 16 bytes per lane ----
typedef int v4i_t __attribute__((ext_vector_type(4)));
typedef __attribute__((address_space(1))) v4i_t* gv4i_p;   // global int4*
typedef __attribute__((address_space(3))) v4i_t* lv4i_p;   // LDS int4*

__device__ __forceinline__ void async_copy16(const float* gsrc, float* ldst) {
#if __has_builtin(__builtin_amdgcn_global_load_async_to_lds_b128)
  __builtin_amdgcn_global_load_async_to_lds_b128(
      (gv4i_p)(uintptr_t)gsrc,
      (lv4i_p)(unsigned)(uintptr_t)ldst,
      0, 0);
#else
  unsigned l = (unsigned)(uintptr_t)ldst;
  unsigned long long g = (unsigned long long)(uintptr_t)gsrc;
  asm volatile("global_load_async_to_lds_b128 %0, %1, off"
               :: "v"(l), "v"(g) : "memory");
#endif
}

#if __has_builtin(__builtin_amdgcn_s_wait_asynccnt)
#define WAIT_ASYNC(n) __builtin_amdgcn_s_wait_asynccnt(n)
#else
#define WAIT_ASYNC(n) asm volatile("s_wait_asynccnt " #n ::: "memory")
#endif

// issue async prefetch of one 1KB probability row into a ring slot (wave 0, 32 lanes)
__device__ __forceinline__ void issue_row(const float* rowbase, float* slotbase, int lane) {
  const float* src = rowbase + lane * 4;   // 16B per lane, lanes 0..31 -> 512B
  float*       dst = slotbase + lane * 4;
  async_copy16(src,       dst);
  async_copy16(src + 128, dst + 128);      // second 512B half
}

__global__ __launch_bounds__(544) void ctc_alpha_kernel(
    const int*  __restrict__ y_true,   // [B, L]
    const float* __restrict__ y_pred,  // [B, T, C]
    float* __restrict__ out)           // [B, 1]
{
  __shared__ float prob[RING * C_];    // ring buffer of prob rows (9 KB)
  __shared__ float alpha[2][S_ + 2];   // ping-pong alpha, 2 NEG pad cells in front

  const int b   = blockIdx.x;
  const int tid = threadIdx.x;         // 0..543; state s = tid for s < 513
  const float* row0 = y_pred + (size_t)b * T_ * C_;

  // ---- wave 0: prime the ring with timesteps 0..DEPTH-1 (slots 0..7) ----
  if (tid < 32) {
    for (int t0 = 0; t0 < DEPTH; ++t0)
      issue_row(row0 + (size_t)t0 * C_, &prob[t0 * C_], tid);
  }

  // ---- per-thread label / can_skip (loads overlap with async prime) ----
  const int s  = tid;
  const int sA = s + 2;                // padded alpha index
  int  lab  = BLANK_;
  bool skip = false;
  if (s < S_ && (s & 1)) {
    lab  = y_true[b * L_ + ((s - 1) >> 1)];
    skip = (s >= 3) && (lab != y_true[b * L_ + ((s - 3) >> 1)]);
  }

  // ---- init: both pads NEG, all of buffer 0 NEG ----
  if (tid < 2) { alpha[0][tid] = NEGV; alpha[1][tid] = NEGV; }
  if (s < S_)  alpha[0][sA] = NEGV;

  // ---- t = 0: wait row 0, seed alpha[0][{0,1}], prefetch row 8 -> slot 8 ----
  if (tid < 32) WAIT_ASYNC(14);        // <= 2*(DEPTH-1): row 0 has landed (in-order)
  __syncthreads();
  if (tid < 32) issue_row(row0 + (size_t)DEPTH * C_, &prob[(RING - 1) * C_], tid);
  if (s < 2) alpha[0][sA] = fast_ln(prob[lab] + 1e-7f);

  // ---- main recurrence: ONE barrier per step ----
  int rd = 1;                          // slot index = t % RING
  for (int t = 1; t < T_; ++t) {
    if (tid < 32) WAIT_ASYNC(14);      // row t guaranteed complete (in-order done)
    __syncthreads();                   // orders: slot-ready, prev alpha writes, prev slot reads

    // prefetch row t+DEPTH into slot (t+DEPTH)%RING == (t-1)%RING (consumed last step)
    if (tid < 32) {
      int tp = t + DEPTH;
      if (tp < T_) {
        int wr = (rd == 0) ? (RING - 1) : (rd - 1);
        issue_row(row0 + (size_t)tp * C_, &prob[wr * C_], tid);
      }
    }

    const float* aOld = alpha[(t + 1) & 1];
    float*       aNew = alpha[t & 1];
    const float* slot = &prob[rd * C_];

    if (s < S_) {
      float a0 = aOld[sA];
      float a1 = aOld[sA - 1];                  // pad makes this valid for s==0
      float a2 = skip ? aOld[sA - 2] : NEGV;
      float m  = fmaxf(a0, fmaxf(a1, a2));
      float sum = fast_exp(a0 - m) + fast_exp(a1 - m) + fast_exp(a2 - m);
      float lp  = fast_ln(slot[lab] + 1e-7f);
      aNew[sA] = m + fast_ln(sum) + lp;
    }

    if (++rd == RING) rd = 0;
  }

  __syncthreads();

  // ---- final: -logsumexp(alpha[S-2], alpha[S-1]) ----
  if (tid == 0) {
    const float* aF = alpha[(T_ - 1) & 1];
    float a = aF[2 + S_ - 2];
    float c = aF[2 + S_ - 1];
    float m = fmaxf(a, c);
    float ll = m + fast_ln(fast_exp(a - m) + fast_exp(c - m));
    out[b] = -ll;
  }
}

extern "C" void kernel_launch(void* const* d_in, const int* in_sizes, int n_in,
                              void* d_out, int out_size, void* d_ws, size_t ws_size,
                              hipStream_t stream) {
  const int*   y_true = (const int*)d_in[0];   // [B, L] int32
  const float* y_pred = (const float*)d_in[1]; // [B, T, C] float32
  float*       out    = (float*)d_out;         // [B, 1] float32
  (void)in_sizes; (void)n_in; (void)out_size; (void)d_ws; (void)ws_size;

  ctc_alpha_kernel<<<dim3(B_), dim3(544), 0, stream>>>(y_true, y_pred, out);
}